// KANLayer_73194832658946
// MI455X (gfx1250) — compile-verified
//
#include <hip/hip_runtime.h>
#include <hip/hip_bf16.h>
#include <stdint.h>

// ---------------------------------------------------------------------------
// KAN layer, MI455X (gfx1250, wave32).
//   out[b,o] = sum_{i,k} sigmoid(bw[i,o,k]*x[b,i] + sc[i,o,k]) * softmax_k(sc[i,o,:]) + bias[o]
// Not a matmul (elementwise nonlinearity in both b- and o-indexed operands),
// so WMMA does not apply; workload is transcendental-VALU bound.
// Optimizations:
//  * sigmoid(z)=0.5*(1+tanh(z/2)) and sum_k softmax = 1  =>  1 trans op/term
//  * fused on-the-fly softmax from the sc vector already in registers
//  * TDM (tensor_load_to_lds, 6-arg clang-23 form) stages the x tile into LDS
//  * coalesced 1KB b128 weight bursts per wave, LDS-broadcast x reads
// ---------------------------------------------------------------------------

#define INPUT_DIM  256
#define OUTPUT_DIM 256
#define NUM_BASIS  8
#define BATCH      512

#define OTILE 32   // outputs per block (one per lane)
#define BTILE 8    // batch rows per block (8 accumulators per lane)
#define NWAVE 4    // waves per block, splitting the i dimension

typedef __attribute__((ext_vector_type(4))) unsigned int tdm_v4u;
typedef __attribute__((ext_vector_type(8))) int          tdm_v8i;
typedef __attribute__((ext_vector_type(4))) int          tdm_v4i;

static __device__ __forceinline__ float fexp2(float a) {
#if __has_builtin(__builtin_amdgcn_exp2f)
  return __builtin_amdgcn_exp2f(a);
#else
  return exp2f(a);
#endif
}

static __device__ __forceinline__ float frcp(float a) {
#if __has_builtin(__builtin_amdgcn_rcpf)
  return __builtin_amdgcn_rcpf(a);
#else
  return 1.0f / a;
#endif
}

// tanh: single transcendental on CDNA5 (V_TANH_F32); exp2-based fallback.
static __device__ __forceinline__ float ftanh(float z) {
#if __has_builtin(__builtin_amdgcn_tanhf)
  return __builtin_amdgcn_tanhf(z);
#elif __has_builtin(__builtin_amdgcn_tanh_f32)
  return __builtin_amdgcn_tanh_f32(z);
#else
  // tanh(z) = 1 - 2/(exp(2z)+1)
  float e = fexp2(z * 2.8853900817779268f); // 2*log2(e)
  return 1.0f - 2.0f * frcp(e + 1.0f);
#endif
}

__global__ __launch_bounds__(NWAVE * 32) void kan_layer_kernel(
    const float* __restrict__ x,     // [BATCH][INPUT_DIM]
    const float* __restrict__ bw,    // [INPUT_DIM][OUTPUT_DIM][NUM_BASIS]
    const float* __restrict__ sc,    // [INPUT_DIM][OUTPUT_DIM][NUM_BASIS]
    const float* __restrict__ bias,  // [OUTPUT_DIM]
    float* __restrict__ out)         // [BATCH][OUTPUT_DIM]
{
  __shared__ float x_s[BTILE * INPUT_DIM];          // 8 KB: x[b0..b0+7][0..255]
  __shared__ float partial[NWAVE][BTILE][OTILE];    // 4 KB: cross-wave reduce

  const int lane = threadIdx.x & 31;
  const int wave = threadIdx.x >> 5;
  const int o0   = blockIdx.x * OTILE;
  const int o    = o0 + lane;
  const int b0   = blockIdx.y * BTILE;

  // ---- Stage x tile (8 rows x 256 floats, contiguous 8KB) into LDS via TDM.
#if __has_builtin(__builtin_amdgcn_tensor_load_to_lds)
  if (wave == 0) {
    unsigned long long ga = (unsigned long long)(const void*)(x + (size_t)b0 * INPUT_DIM);
    unsigned lds_off = (unsigned)(unsigned long long)(void*)&x_s[0]; // LDS byte offset

    // D# group 0 (ISA 8.3): count=1 | lds_addr | global_addr[56:0] | type=2
    tdm_v4u g0;
    g0[0] = 1u;
    g0[1] = lds_off;
    g0[2] = (unsigned)(ga & 0xFFFFFFFFull);
    g0[3] = (unsigned)((ga >> 32) & 0x01FFFFFFull) | (2u << 30);

    // D# group 1 (ISA 8.4): data_size=2 (4B); 1 row of 2048 elements.
    tdm_v8i g1;
    g1[0] = (2 << 16);                    // wg_mask=0, data_size=4B, no iter/pad
    g1[1] = (int)(2048u << 16);           // tensor_dim0[15:0] = 2048
    g1[2] = (int)(1u << 16);              // tensor_dim0 hi=0 | tensor_dim1=1
    g1[3] = (int)(2048u << 16);           // tensor_dim1 hi=0 | tile_dim0=2048
    g1[4] = 1;                            // tile_dim1=1, tile_dim2=0
    g1[5] = 2048;                         // tensor_dim0_stride lo32
    g1[6] = 0;                            // stride hi | tensor_dim1_stride lo
    g1[7] = 0;

    tdm_v4i gz4 = {0, 0, 0, 0};           // groups 2/3 unused (2D tile)
    tdm_v8i gz8 = {0, 0, 0, 0, 0, 0, 0, 0};
    // clang-23 / therock-10.0 6-arg form:
    //   (v4u g0, v8i g1, v4i g2, v4i g3, v8i extra, i32 cpol)
    __builtin_amdgcn_tensor_load_to_lds(g0, g1, gz4, gz4, gz8, 0);
  }
  __builtin_amdgcn_s_wait_tensorcnt(0);
  __syncthreads();
#else
  for (int t = threadIdx.x; t < BTILE * INPUT_DIM; t += blockDim.x)
    x_s[t] = x[(size_t)b0 * INPUT_DIM + t];
  __syncthreads();
#endif

  const float LOG2E = 1.4426950408889634f;
  const int i0 = wave * (INPUT_DIM / NWAVE);
  const int i1 = i0 + (INPUT_DIM / NWAVE);

  float acc[BTILE];
#pragma unroll
  for (int j = 0; j < BTILE; ++j) acc[j] = 0.0f;

  for (int i = i0; i < i1; ++i) {
    const size_t base = ((size_t)i * OUTPUT_DIM + o) * NUM_BASIS;
    // 32 lanes * 32B each = fully coalesced 1KB bursts (b128 x2 per array)
    float4 w0 = *(const float4*)(bw + base);
    float4 w1 = *(const float4*)(bw + base + 4);
    float4 c0 = *(const float4*)(sc + base);
    float4 c1 = *(const float4*)(sc + base + 4);

    float w[NUM_BASIS] = {w0.x, w0.y, w0.z, w0.w, w1.x, w1.y, w1.z, w1.w};
    float c[NUM_BASIS] = {c0.x, c0.y, c0.z, c0.w, c1.x, c1.y, c1.z, c1.w};

    // softmax numerators e_k = exp(sc_k); values ~N(0, 0.02^2): no max-shift needed
    float e[NUM_BASIS];
    float esum = 0.0f;
#pragma unroll
    for (int k = 0; k < NUM_BASIS; ++k) {
      e[k] = fexp2(c[k] * LOG2E);
      esum += e[k];
    }
    const float r = frcp(esum);

    // pre-halve for the tanh identity (amortized over BTILE batch rows)
    float bh[NUM_BASIS], sh[NUM_BASIS];
#pragma unroll
    for (int k = 0; k < NUM_BASIS; ++k) { bh[k] = 0.5f * w[k]; sh[k] = 0.5f * c[k]; }

#pragma unroll
    for (int j = 0; j < BTILE; ++j) {
      const float xb = x_s[j * INPUT_DIM + i];   // wave-uniform: LDS broadcast
      float s = 0.0f;
#pragma unroll
      for (int k = 0; k < NUM_BASIS; ++k) {
        float z = __builtin_fmaf(bh[k], xb, sh[k]);  // (bw*x + sc)/2
        float t = ftanh(z);
        s = __builtin_fmaf(e[k], t, s);
      }
      acc[j] = __builtin_fmaf(s, r, acc[j]);     // += sum_k p_k * tanh(...)
    }
  }

  // ---- cross-wave (i-split) reduction through LDS
#pragma unroll
  for (int j = 0; j < BTILE; ++j) partial[wave][j][lane] = acc[j];
  __syncthreads();

  for (int idx = threadIdx.x; idx < BTILE * OTILE; idx += blockDim.x) {
    const int bl = idx >> 5;        // 0..7
    const int ol = idx & 31;        // 0..31
    float s = partial[0][bl][ol] + partial[1][bl][ol] +
              partial[2][bl][ol] + partial[3][bl][ol];
    // sum_i sum_k p_k*sigmoid = 0.5*INPUT_DIM + 0.5 * s
    out[(size_t)(b0 + bl) * OUTPUT_DIM + (o0 + ol)] =
        0.5f * (float)INPUT_DIM + 0.5f * s + bias[o0 + ol];
  }
}

extern "C" void kernel_launch(void* const* d_in, const int* in_sizes, int n_in,
                              void* d_out, int out_size, void* d_ws, size_t ws_size,
                              hipStream_t stream) {
  (void)in_sizes; (void)n_in; (void)out_size; (void)d_ws; (void)ws_size;
  const float* x    = (const float*)d_in[0];
  const float* bw   = (const float*)d_in[1];
  const float* sc   = (const float*)d_in[2];
  const float* bias = (const float*)d_in[3];
  float* out = (float*)d_out;

  dim3 grid(OUTPUT_DIM / OTILE, BATCH / BTILE);  // 8 x 64 = 512 blocks
  dim3 block(NWAVE * 32);                        // 128 threads = 4 waves
  kan_layer_kernel<<<grid, block, 0, stream>>>(x, bw, sc, bias, out);
}